// NRRenderer_73993696575488
// MI455X (gfx1250) — compile-verified
//
#include <hip/hip_runtime.h>

// ---------------------------------------------------------------------------
// Neural-renderer forward (rasterize + z-buffer + SSE loss) for gfx1250.
// Core rasterization mapped onto V_WMMA_F32_16X16X4_F32:
//   D[16 pix][16 faces] = A[16 pix][(x,y,1,0)] x B[(cx,cy,cc,0)][16 faces]
// Four WMMAs per face-group compute b0, b1, b2, z simultaneously.
// B panel is staged per-block into LDS with double-buffered
// GLOBAL_LOAD_ASYNC_TO_LDS_B128 (ASYNCcnt / s_wait_asynccnt).
// ---------------------------------------------------------------------------

typedef float v2f __attribute__((ext_vector_type(2)));
typedef float v8f __attribute__((ext_vector_type(8)));

#define HH 512
#define WW 512
#define PP (HH * WW)
#define VIEW_TAN_F 0.57735026918962576f
#define NEAR_F 0.1f
#define BIG_F 1.0e10f

#define GROUPS_PER_CHUNK 8                       // 8 groups * 1KB = 8KB / chunk
#define CHUNK_FLOAT4 512                         // 8KB in float4

static __device__ inline v2f mk2(float x, float y) { v2f r; r.x = x; r.y = y; return r; }

// ---------------------------------------------------------------------------
// Kernel 1: transform + project vertices (camera basis recomputed per thread).
// ---------------------------------------------------------------------------
__global__ void k_vertices(const float* __restrict__ verts,
                           const float* __restrict__ campos,
                           const float* __restrict__ camup,
                           float* __restrict__ projx,
                           float* __restrict__ projy,
                           float* __restrict__ zca,
                           int V) {
  int i = blockIdx.x * blockDim.x + threadIdx.x;
  if (i >= V) return;

  float ex = campos[0], ey = campos[1], ez = campos[2];
  float zx = -ex, zy = -ey, zz = -ez;
  float inv = 1.0f / (sqrtf(zx * zx + zy * zy + zz * zz) + 1e-12f);
  zx *= inv; zy *= inv; zz *= inv;
  float ux = camup[0], uy = camup[1], uz = camup[2];
  inv = 1.0f / (sqrtf(ux * ux + uy * uy + uz * uz) + 1e-12f);
  ux *= inv; uy *= inv; uz *= inv;
  float xx = uy * zz - uz * zy;
  float xy = uz * zx - ux * zz;
  float xz = ux * zy - uy * zx;
  inv = 1.0f / (sqrtf(xx * xx + xy * xy + xz * xz) + 1e-12f);
  xx *= inv; xy *= inv; xz *= inv;
  float yx = zy * xz - zz * xy;
  float yy = zz * xx - zx * xz;
  float yz = zx * xy - zy * xx;

  float vx = verts[i * 3 + 0] - ex;
  float vy = verts[i * 3 + 1] - ey;
  float vz = verts[i * 3 + 2] - ez;
  float cx = xx * vx + xy * vy + xz * vz;
  float cy = yx * vx + yy * vy + yz * vz;
  float cz = zx * vx + zy * vy + zz * vz;
  float denom = cz * VIEW_TAN_F + 1e-12f;
  projx[i] = cx / denom;
  projy[i] = cy / denom;
  zca[i]   = cz;
}

// ---------------------------------------------------------------------------
// Kernel 2: per-face setup. Fold sign(area) and 1/area into plane coeffs:
// b_i(p) = bix*px + biy*py + bic, inside <=> all b_i >= 0,
// z(p) = zx*px + zy*py + zc. Packed into WMMA 4x16 B layout:
//   lanes 0..15 -> (cx, cy)  (K=0,1) ; lanes 16..31 -> (cc, 0)  (K=2,3)
// ---------------------------------------------------------------------------
__global__ void k_faces(const int* __restrict__ faces,
                        const float* __restrict__ projx,
                        const float* __restrict__ projy,
                        const float* __restrict__ zca,
                        const float* __restrict__ tex,
                        float* __restrict__ packedB,
                        float* __restrict__ cols,
                        int F) {
  int f = blockIdx.x * blockDim.x + threadIdx.x;
  if (f >= F) return;

  int i0 = faces[f * 3 + 0];
  int i1 = faces[f * 3 + 1];
  int i2 = faces[f * 3 + 2];
  float p0x = projx[i0], p0y = projy[i0], z0 = zca[i0];
  float p1x = projx[i1], p1y = projy[i1], z1 = zca[i1];
  float p2x = projx[i2], p2y = projy[i2], z2 = zca[i2];

  float e1x = p1x - p0x, e1y = p1y - p0y;
  float e2x = p2x - p0x, e2y = p2y - p0y;
  float area = e1x * e2y - e1y * e2x;
  bool valid = fabsf(area) > 1e-8f;

  float b0x, b0y, b0c, b1x, b1y, b1c, b2x, b2y, b2c, zpx, zpy, zpc;
  if (valid) {
    float inv = 1.0f / area;                 // == sign(area)/|area|
    float ex, ey;
    ex = p2x - p1x; ey = p2y - p1y;          // w0: edge p1->p2
    b0x = -ey * inv; b0y = ex * inv; b0c = (ey * p1x - ex * p1y) * inv;
    ex = p0x - p2x; ey = p0y - p2y;          // w1: edge p2->p0
    b1x = -ey * inv; b1y = ex * inv; b1c = (ey * p2x - ex * p2y) * inv;
    ex = p1x - p0x; ey = p1y - p0y;          // w2: edge p0->p1
    b2x = -ey * inv; b2y = ex * inv; b2c = (ey * p0x - ex * p0y) * inv;
    zpx = z0 * b0x + z1 * b1x + z2 * b2x;
    zpy = z0 * b0y + z1 * b1y + z2 * b2y;
    zpc = z0 * b0c + z1 * b1c + z2 * b2c;
  } else {
    b0x = b0y = 0.0f; b0c = -1.0f;           // never inside
    b1x = b1y = 0.0f; b1c = -1.0f;
    b2x = b2y = 0.0f; b2c = -1.0f;
    zpx = zpy = zpc = 0.0f;
  }

  int g = f >> 4;
  int n = f & 15;
  v2f* B = reinterpret_cast<v2f*>(packedB) + (size_t)g * 128;
  B[0 * 32 + n]      = mk2(b0x, b0y);
  B[0 * 32 + 16 + n] = mk2(b0c, 0.0f);
  B[1 * 32 + n]      = mk2(b1x, b1y);
  B[1 * 32 + 16 + n] = mk2(b1c, 0.0f);
  B[2 * 32 + n]      = mk2(b2x, b2y);
  B[2 * 32 + 16 + n] = mk2(b2c, 0.0f);
  B[3 * 32 + n]      = mk2(zpx, zpy);
  B[3 * 32 + 16 + n] = mk2(zpc, 0.0f);

  const float* t = tex + (size_t)f * 24;     // (2,2,2,3) texels
  float c0 = 0.0f, c1 = 0.0f, c2 = 0.0f;
#pragma unroll
  for (int j = 0; j < 8; ++j) {
    c0 += t[j * 3 + 0];
    c1 += t[j * 3 + 1];
    c2 += t[j * 3 + 2];
  }
  cols[f * 3 + 0] = c0 * 0.125f;
  cols[f * 3 + 1] = c1 * 0.125f;
  cols[f * 3 + 2] = c2 * 0.125f;
}

// ---------------------------------------------------------------------------
// Kernel 3: WMMA rasterizer with async double-buffered LDS staging of B.
// One wave owns 16 pixels; block stages 8 face-groups (8KB) per chunk via
// global_load_async_to_lds_b128 while computing the previous chunk.
// ---------------------------------------------------------------------------
__global__ void __launch_bounds__(256)
k_raster(const float* __restrict__ packedB,
         const float* __restrict__ cols,
         const float* __restrict__ ref,
         float* __restrict__ waveLoss,
         int ngroups) {
  __shared__ float4   bufB[2][CHUNK_FLOAT4];   // 2 x 8KB B-panel staging
  __shared__ float    sZ[8][16][16];
  __shared__ unsigned sI[8][16][16];

  const int tid  = threadIdx.x;
  const int lane = tid & 31;
  const int wIn  = tid >> 5;
  const int wave = blockIdx.x * 8 + wIn;
  const int pixBase = wave * 16;
  const int col  = lane & 15;
  const int half = lane >> 4;
  const int nchunks = ngroups / GROUPS_PER_CHUNK;

  // A matrix (16x4 f32 layout): lanes 0..15 carry (px,py) for pixel M=lane,
  // lanes 16..31 carry (1,0) for K=2,3 (affine constant term).
  v2f a;
  if (lane < 16) {
    int p  = pixBase + lane;
    int ix = p & (WW - 1);
    int iy = p >> 9;                              // W == 512
    a = mk2((ix + 0.5f) * (2.0f / WW) - 1.0f,
            1.0f - (iy + 0.5f) * (2.0f / HH));
  } else {
    a = mk2(1.0f, 0.0f);
  }

  float    zmin[8];
  unsigned fidx[8];
#pragma unroll
  for (int r = 0; r < 8; ++r) { zmin[r] = BIG_F; fidx[r] = 0xFFFFFFFFu; }

  // Per-chunk async stage: 512 float4 by 256 threads -> 2 B128 DMAs each.
  // NOTE: inst offset is added to BOTH the LDS and the global address
  // (ISA 10.7/15.18 async pseudocode), so offset:4096 yields element t+256
  // on both sides.
  const unsigned ldsBase0 = (unsigned)(size_t)(&bufB[0][0]);
  const unsigned ldsBase1 = (unsigned)(size_t)(&bufB[1][0]);
  const unsigned voff     = (unsigned)tid * 16u;

  // Prologue: kick off chunk 0 into buffer 0.
  {
    const float* gsrc = packedB;                 // chunk 0
    asm volatile("global_load_async_to_lds_b128 %0, %1, %2\n\t"
                 "global_load_async_to_lds_b128 %0, %1, %2 offset:4096"
                 :: "v"(ldsBase0 + voff), "v"(voff), "s"(gsrc)
                 : "memory");
  }

  for (int c = 0; c < nchunks; ++c) {
    if (c + 1 < nchunks) {
      // Stage next chunk into the other buffer, then retire current chunk's
      // two DMAs (leave next chunk's two in flight).
      const float* gsrc = packedB + (size_t)(c + 1) * (CHUNK_FLOAT4 * 4);
      const unsigned ldsNext = ((c + 1) & 1) ? ldsBase1 : ldsBase0;
      asm volatile("global_load_async_to_lds_b128 %0, %1, %2\n\t"
                   "global_load_async_to_lds_b128 %0, %1, %2 offset:4096"
                   :: "v"(ldsNext + voff), "v"(voff), "s"(gsrc)
                   : "memory");
      asm volatile("s_wait_asynccnt 0x2" ::: "memory");
    } else {
      asm volatile("s_wait_asynccnt 0x0" ::: "memory");
    }
    __syncthreads();                             // all waves' DMA visible

    const v2f* Lb = reinterpret_cast<const v2f*>(&bufB[c & 1][0]);
    for (int gin = 0; gin < GROUPS_PER_CHUNK; ++gin) {
      const v2f* Bg = Lb + gin * 128 + lane;     // ds_load_b64 x4
      v2f b0 = Bg[0];
      v2f b1 = Bg[32];
      v2f b2 = Bg[64];
      v2f bz = Bg[96];

      v8f zero = {};
      v8f d0 = __builtin_amdgcn_wmma_f32_16x16x4_f32(false, a, false, b0, (short)0, zero, false, false);
      v8f d1 = __builtin_amdgcn_wmma_f32_16x16x4_f32(false, a, false, b1, (short)0, zero, false, false);
      v8f d2 = __builtin_amdgcn_wmma_f32_16x16x4_f32(false, a, false, b2, (short)0, zero, false, false);
      v8f dz = __builtin_amdgcn_wmma_f32_16x16x4_f32(false, a, false, bz, (short)0, zero, false, false);

      unsigned face = (unsigned)((c * GROUPS_PER_CHUNK + gin) * 16 + col);
#pragma unroll
      for (int r = 0; r < 8; ++r) {
        float m = fminf(fminf(d0[r], d1[r]), d2[r]);
        float z = dz[r];
        float zm = ((m >= 0.0f) && (z > NEAR_F)) ? z : BIG_F;
        if (zm < zmin[r]) { zmin[r] = zm; fidx[r] = face; }
      }
    }
    __syncthreads();                             // done reading buf[c&1]
  }

#pragma unroll
  for (int r = 0; r < 8; ++r) {
    sZ[wIn][half * 8 + r][col] = zmin[r];
    sI[wIn][half * 8 + r][col] = fidx[r];
  }
  __syncthreads();

  float loss = 0.0f;
  if (lane < 16) {
    float    bzv = BIG_F;
    unsigned bi  = 0xFFFFFFFFu;
#pragma unroll
    for (int cc = 0; cc < 16; ++cc) {
      float    z = sZ[wIn][lane][cc];
      unsigned i = sI[wIn][lane][cc];
      if (z < bzv || (z == bzv && i < bi)) { bzv = z; bi = i; }
    }
    float c0 = 0.0f, c1 = 0.0f, c2 = 0.0f;
    if (bi != 0xFFFFFFFFu) {
      c0 = cols[bi * 3 + 0];
      c1 = cols[bi * 3 + 1];
      c2 = cols[bi * 3 + 2];
    }
    int p = pixBase + lane;
    // flipped[c] = orig[2-c]  (image[:, ::-1] flips the channel axis)
    float e0 = c2 - ref[0 * PP + p];
    float e1 = c1 - ref[1 * PP + p];
    float e2 = c0 - ref[2 * PP + p];
    loss = e0 * e0 + e1 * e1 + e2 * e2;
  }
#pragma unroll
  for (int m = 16; m >= 1; m >>= 1) loss += __shfl_xor(loss, m, 32);
  if (lane == 0) waveLoss[wave] = loss;
}

// ---------------------------------------------------------------------------
// Kernel 4: deterministic reduction of per-wave losses to the scalar.
// ---------------------------------------------------------------------------
__global__ void k_reduce(const float* __restrict__ in, float* __restrict__ out, int n) {
  __shared__ float s[256];
  float acc = 0.0f;
  for (int i = threadIdx.x; i < n; i += 256) acc += in[i];
  s[threadIdx.x] = acc;
  __syncthreads();
  for (int st = 128; st > 0; st >>= 1) {
    if ((int)threadIdx.x < st) s[threadIdx.x] += s[threadIdx.x + st];
    __syncthreads();
  }
  if (threadIdx.x == 0) out[0] = s[0];
}

// ---------------------------------------------------------------------------
extern "C" void kernel_launch(void* const* d_in, const int* in_sizes, int n_in,
                              void* d_out, int out_size, void* d_ws, size_t ws_size,
                              hipStream_t stream) {
  const float* verts  = (const float*)d_in[0];
  const int*   faces  = (const int*)d_in[1];
  const float* tex    = (const float*)d_in[2];
  const float* campos = (const float*)d_in[3];
  const float* camup  = (const float*)d_in[4];
  const float* ref    = (const float*)d_in[5];
  float*       out    = (float*)d_out;
  float*       ws     = (float*)d_ws;

  const int V = in_sizes[0] / 3;     // 2048
  const int F = in_sizes[1] / 3;     // 4096
  const int ngroups = F / 16;        // 256
  const int nWaves  = PP / 16;       // 16384

  // Workspace layout (floats)
  float* projx    = ws;
  float* projy    = projx + V;
  float* zca      = projy + V;
  float* packedB  = zca + V;                     // ngroups*4*32*2 floats
  float* colsBuf  = packedB + (size_t)ngroups * 256;
  float* waveLoss = colsBuf + (size_t)F * 3;

  k_vertices<<<(V + 255) / 256, 256, 0, stream>>>(verts, campos, camup,
                                                  projx, projy, zca, V);
  k_faces<<<(F + 255) / 256, 256, 0, stream>>>(faces, projx, projy, zca, tex,
                                               packedB, colsBuf, F);
  k_raster<<<nWaves / 8, 256, 0, stream>>>(packedB, colsBuf, ref, waveLoss, ngroups);
  k_reduce<<<1, 256, 0, stream>>>(waveLoss, out, nWaves);
}